// LeniaStepConv_25898652795134
// MI455X (gfx1250) — compile-verified
//
#include <hip/hip_runtime.h>
#include <math.h>

// ---------------------------------------------------------------------------
// Lenia step, MI455X (gfx1250). Conv-as-GEMM with V_WMMA_F32_16X16X4_F32 and
// async global->LDS staging (gfx1250 ASYNCcnt path).
// Grid 128x128, R=20 -> 41x41 periodic convolution per rule (4 rules).
// ---------------------------------------------------------------------------

#define G 128
#define NCELL (G * G)
#define R 20
#define D 41                  // 2R+1
#define NSHIFT (D * D)        // 1681
#define NBVEC 1850            // 1 (self) + (2R+3)^2
#define NRULES 4
#define WPADW 72              // padded kernel row: 15 zeros | 41 taps | 16 zeros
#define TINV 0.1f             // 1/T

typedef __attribute__((ext_vector_type(2))) float v2f;
typedef __attribute__((ext_vector_type(8))) float v8f;

// ---- gfx1250 async global->LDS helpers (tracked by ASYNCcnt) --------------
__device__ __forceinline__ void async_to_lds_b32(uint32_t lds_off, const void* gp) {
    asm volatile("global_load_async_to_lds_b32 %0, %1, off"
                 :: "v"(lds_off), "v"(gp) : "memory");
}
__device__ __forceinline__ void async_to_lds_b128(uint32_t lds_off, const void* gp) {
    asm volatile("global_load_async_to_lds_b128 %0, %1, off"
                 :: "v"(lds_off), "v"(gp) : "memory");
}
__device__ __forceinline__ void wait_asynccnt0() {
    asm volatile("s_wait_asynccnt 0x0" ::: "memory");
}

// ---------------------------------------------------------------------------
// Kernel 1: build the 4 normalized conv kernels (zero-padded rows) + padsums.
// One block, 256 threads. Writes Wpad[4][41][72] and padsum[4] to workspace.
// ---------------------------------------------------------------------------
__global__ __launch_bounds__(256)
void lenia_weights_kernel(const float* __restrict__ r,
                          const float* __restrict__ rk,
                          const float* __restrict__ b,
                          const float* __restrict__ w,
                          float* __restrict__ Wpad,    // [NRULES][D][WPADW]
                          float* __restrict__ padsum)  // [NRULES]
{
    __shared__ float wk_s[NBVEC];
    __shared__ float red[256];
    const int t = threadIdx.x;

    // zero the padded kernel table once
    for (int e = t; e < NRULES * D * WPADW; e += 256) Wpad[e] = 0.f;
    __syncthreads();

    for (int k = 0; k < NRULES; ++k) {
        const float rkk = r[k];
        // ---- raw (unnormalized) kernel values over the 1850-slot vector ----
        for (int j = t; j < NBVEC; j += 256) {
            float dist;
            if (j == 0) {
                dist = 0.f;                       // self slot
            } else if (j <= NSHIFT) {
                const int s  = j - 1;
                const int si = s / D - R;
                const int sj = s % D - R;
                dist = sqrtf((float)(si * si + sj * sj)) * (1.0f / (float)R);
            } else {
                // padding slots gather cell 0 from the center cell (64,64)
                dist = sqrtf(8192.f) * (1.0f / (float)R);   // sqrt(64^2+64^2)/R
            }
            const float edge = 1.f / (1.f + expf((dist - 1.f) * 10.f));
            float acc = 0.f;
            for (int c = 0; c < 3; ++c) {
                const float z = (dist / rkk - rk[k * 3 + c]) / w[k * 3 + c];
                acc += b[k * 3 + c] * expf(-0.5f * z * z);
            }
            wk_s[j] = edge * acc;
        }
        __syncthreads();

        // ---- total sum (normalization) ----
        float ls = 0.f;
        for (int j = t; j < NBVEC; j += 256) ls += wk_s[j];
        red[t] = ls;
        __syncthreads();
        for (int o = 128; o > 0; o >>= 1) {
            if (t < o) red[t] += red[t + o];
            __syncthreads();
        }
        const float inv = (1.f / red[0]) * (1.f / (float)NBVEC); // wn/1850
        __syncthreads();

        // ---- padding-slot sum ----
        float ps = 0.f;
        for (int j = NSHIFT + 1 + t; j < NBVEC; j += 256) ps += wk_s[j];
        red[t] = ps;
        __syncthreads();
        for (int o = 128; o > 0; o >>= 1) {
            if (t < o) red[t] += red[t + o];
            __syncthreads();
        }
        if (t == 0) padsum[k] = red[0] * inv;

        // ---- write normalized taps into padded rows; fold self slot in ----
        for (int s = t; s < NSHIFT; s += 256) {
            float val = wk_s[1 + s] * inv;
            if (s == R * D + R) val += wk_s[0] * inv;  // (0,0) tap doubled
            const int si = s / D, sj = s % D;
            Wpad[(k * D + si) * WPADW + 15 + sj] = val;
        }
        __syncthreads();   // wk_s / red reused next rule
    }
}

// ---------------------------------------------------------------------------
// Kernel 2: copy pos -> d_out[0 : 2N]  (tuple return order: pos first)
// ---------------------------------------------------------------------------
__global__ __launch_bounds__(256)
void lenia_copy_pos_kernel(const float* __restrict__ pos, float* __restrict__ out)
{
    const int i = blockIdx.x * 256 + threadIdx.x;
    if (i < 2 * NCELL) out[i] = pos[i];
}

// ---------------------------------------------------------------------------
// Kernel 3: convolution + growth update. One block per 16x16 output tile,
// 4 waves per block (one rule per wave). WMMA f32 16x16x4:
//   out[m,n] += sum_si A_si(16x56) x B_si(56x16 Toeplitz of kernel row si)
// B rows come from the zero-padded kernel rows -> unconditional LDS loads.
// ---------------------------------------------------------------------------
__global__ __launch_bounds__(128)
void lenia_conv_kernel(const float* __restrict__ x,
                       const int*   __restrict__ c0,
                       const int*   __restrict__ c1,
                       const float* __restrict__ h,
                       const float* __restrict__ m,
                       const float* __restrict__ s,
                       const float* __restrict__ Wpad,
                       const float* __restrict__ padsum,
                       float* __restrict__ out_x)
{
    __shared__ float xp[3][56][57];                       // wrapped input patch
    __shared__ __align__(16) float wlp[NRULES][D][WPADW]; // padded kernel rows
    __shared__ float dacc[16][16][3];                     // per-cell deltas

    const int tid = threadIdx.x;
    const int ti = blockIdx.x >> 3, tj = blockIdx.x & 7;
    const int i0 = ti * 16, j0 = tj * 16;

    // ---- async-stage patch (periodic wrap, per-lane gather) ----
    for (int e = tid; e < 3 * 56 * 56; e += 128) {
        const int c  = e / (56 * 56);
        const int rr = (e / 56) % 56;
        const int cc = e % 56;
        const int gi = (i0 - R + rr) & (G - 1);
        const int gj = (j0 - R + cc) & (G - 1);
        async_to_lds_b32((uint32_t)(uintptr_t)&xp[c][rr][cc],
                         x + ((gi * G + gj) * 3 + c));
    }
    // ---- async-stage padded kernel table (contiguous, 16B chunks) ----
    {
        float* wflat = &wlp[0][0][0];
        const int nvec = (NRULES * D * WPADW) / 4;   // 2952 float4 chunks
        for (int e = tid; e < nvec; e += 128) {
            async_to_lds_b128((uint32_t)(uintptr_t)(wflat + e * 4),
                              Wpad + e * 4);
        }
    }
    for (int e = tid; e < 16 * 16 * 3; e += 128)
        (&dacc[0][0][0])[e] = 0.f;
    wait_asynccnt0();
    __syncthreads();

    const int wave  = tid >> 5;      // rule index 0..3
    const int lane  = tid & 31;
    const int half  = lane >> 4;     // lane-half selects K {0,1}/{2,3} split
    const int mn    = lane & 15;     // A: row M; B: col N; D: col N
    const int ch    = c0[wave];

    v8f acc = {};
    for (int si = 0; si < D; ++si) {
        // B base shifted so wr[kk] = Wrow[kk - mn] with zero padding built in
        const float* __restrict__ wr   = &wlp[wave][si][15 - mn];
        const float* __restrict__ arow = &xp[ch][si + mn][0];
        #pragma unroll
        for (int t = 0; t < 14; ++t) {
            const int kk = 4 * t + 2 * half;
            v2f a, bb;
            a.x  = arow[kk];
            a.y  = arow[kk + 1];
            bb.x = wr[kk];          // Toeplitz: B[p][n] = W[si][p-n], padded
            bb.y = wr[kk + 1];
            acc = __builtin_amdgcn_wmma_f32_16x16x4_f32(
                      /*neg_a=*/false, a, /*neg_b=*/false, bb,
                      /*c_mod=*/(short)0, acc,
                      /*reuse_a=*/false, /*reuse_b=*/false);
        }
    }

    // ---- growth field per rule; combine rules via LDS float atomics ----
    const float x0c  = x[ch];                 // cell 0, channel c0[k]
    const float padc = padsum[wave] * x0c;    // padding-slot contribution (~0)
    const float mk = m[wave], sk = s[wave], hk = h[wave];
    const int   tc = c1[wave];
    #pragma unroll
    for (int d = 0; d < 8; ++d) {
        const float p = acc[d] + padc;
        const float z = p - mk;
        const float field = expf(-z * z / (2.f * sk * sk) - 0.001f) * 2.f - 1.f;
        atomicAdd(&dacc[d + 8 * half][mn][tc], hk * field);
    }
    __syncthreads();

    // ---- clipped update, store new_x ----
    for (int e = tid; e < 256; e += 128) {
        const int rr = e >> 4, cc = e & 15;
        const int cell = (i0 + rr) * G + (j0 + cc);
        #pragma unroll
        for (int c = 0; c < 3; ++c) {
            float nv = x[cell * 3 + c] + dacc[rr][cc][c] * TINV;
            nv = fminf(fmaxf(nv, 0.f), 1.f);
            out_x[cell * 3 + c] = nv;
        }
    }
}

// ---------------------------------------------------------------------------
extern "C" void kernel_launch(void* const* d_in, const int* in_sizes, int n_in,
                              void* d_out, int out_size, void* d_ws, size_t ws_size,
                              hipStream_t stream)
{
    const float* pos = (const float*)d_in[0];
    const float* x   = (const float*)d_in[1];
    const int*   c0  = (const int*)  d_in[2];
    const int*   c1  = (const int*)  d_in[3];
    const float* r   = (const float*)d_in[4];
    const float* rk  = (const float*)d_in[5];
    const float* b   = (const float*)d_in[6];
    const float* w   = (const float*)d_in[7];
    const float* h   = (const float*)d_in[8];
    const float* m   = (const float*)d_in[9];
    const float* s   = (const float*)d_in[10];

    float* out  = (float*)d_out;              // [pos (2N) | new_x (3N)]
    float* Wp   = (float*)d_ws;               // [4][41][72] padded kernels
    float* ps   = Wp + NRULES * D * WPADW;    // [4]

    lenia_weights_kernel<<<1, 256, 0, stream>>>(r, rk, b, w, Wp, ps);
    lenia_copy_pos_kernel<<<(2 * NCELL + 255) / 256, 256, 0, stream>>>(pos, out);
    lenia_conv_kernel<<<64, 128, 0, stream>>>(x, c0, c1, h, m, s, Wp, ps,
                                              out + 2 * NCELL);
}